// PortfolioLoss_16621523435816
// MI455X (gfx1250) — compile-verified
//
#include <hip/hip_runtime.h>
#include <stdint.h>

// Problem shape (fixed by the reference)
constexpr int BB = 4096;
constexpr int NN = 8192;
constexpr float EPSF = 1e-8f;

// Workspace layout (floats)
constexpr size_t WS_INV  = 0;                 // [BB] 1/(sum_w+EPS)
constexpr size_t WS_ENT  = WS_INV + BB;       // [BB] sum w*log(w+EPS)
constexpr size_t WS_SC   = WS_ENT + BB;       // [BB] sum w*c*m
constexpr size_t WS_TW   = WS_SC  + BB;       // [BB] sum w*m
constexpr size_t WS_RK   = WS_TW  + BB;       // [BB] ranking contribution (gated, /n_valid)
constexpr size_t WS_OK   = WS_RK  + BB;       // [BB] ok flag
constexpr size_t WS_PART = WS_OK  + BB;       // [32*NN] column partials
constexpr size_t WS_COL  = WS_PART + 32ull*NN;// [NN] colsum

typedef __attribute__((ext_vector_type(2))) float v2f;
typedef __attribute__((ext_vector_type(8))) float v8f;

// ---------------------------------------------------------------------------
// Kernel 1: per-row reductions. One block per row, 256 threads, async
// double-buffered global->LDS staging of 5 arrays (gfx1250 async path).
// Single-use streams (conf / rets / scores) are loaded non-temporal so the
// weights+masks lines survive in L2 for the column pass.
// ---------------------------------------------------------------------------
__global__ __launch_bounds__(256) void row_kernel(
    const float* __restrict__ weights, const float* __restrict__ conf,
    const float* __restrict__ rets,    const float* __restrict__ masks,
    const float* __restrict__ scores,  float* __restrict__ ws)
{
  constexpr int TILE = 1024;         // floats per tile per array (4 KB)
  constexpr int NT   = NN / TILE;    // 8 tiles

  const int b   = blockIdx.x;
  const int tid = threadIdx.x;

  __shared__ float4 stage[2][5][256];   // 40 KB staging (double buffer x 5 arrays)
  __shared__ float  red[10][256];       // 10 KB reduction scratch

  const size_t rowOff = (size_t)b * NN;
  const float* gptr[5] = { weights + rowOff, conf + rowOff, rets + rowOff,
                           masks + rowOff, scores + rowOff };

  // Low 32 bits of a flat pointer to __shared__ == group-segment byte offset.
  const uint32_t ldsBase = (uint32_t)(uintptr_t)&stage[0][0][0];

  auto issue_tile = [&](int buf, int tile) {
    const int tbase = tile * TILE + tid * 4;
#pragma unroll
    for (int a = 0; a < 5; ++a) {
      uint32_t dst = ldsBase + (uint32_t)(((buf * 5 + a) * 256 + tid) * 16);
      const float* src = gptr[a] + tbase;
      if (a == 0 || a == 3) {
        // weights / masks: re-read by the column pass -> keep cacheable
        asm volatile("global_load_async_to_lds_b128 %0, %1, off"
                     :: "v"(dst), "v"(src) : "memory");
      } else {
        // conf / rets / scores: single-use stream -> non-temporal
        asm volatile("global_load_async_to_lds_b128 %0, %1, off th:TH_LOAD_NT"
                     :: "v"(dst), "v"(src) : "memory");
      }
    }
  };

  // Accumulators
  float sum_w = 0.f, ent = 0.f, sc = 0.f, tw = 0.f, nv = 0.f;
  // online softmax state for target (= returns*20) with cross-sum of scores
  float mt = -__builtin_inff(), zt = 0.f, ct = 0.f;
  // online softmax state for scores
  float ms = -__builtin_inff(), zs = 0.f;

  auto proc = [&](float w, float c, float r, float m, float s) {
    sum_w += w;
    ent   += w * __logf(w + EPSF);
    float wm = w * m;
    sc += wm * c;
    tw += wm;
    nv += m;
    if (m > 0.f) {
      float t = r * 20.f;                 // RETURN_TEMP
      if (t > mt) {                        // rescale: exp(-inf)=0 handles init
        float k = __expf(mt - t);
        zt = zt * k + 1.f;
        ct = ct * k + s;
        mt = t;
      } else {
        float e = __expf(t - mt);
        zt += e;
        ct += e * s;
      }
      if (s > ms) {
        float k = __expf(ms - s);
        zs = zs * k + 1.f;
        ms = s;
      } else {
        zs += __expf(s - ms);
      }
    }
  };

  issue_tile(0, 0);
  for (int t = 0; t < NT; ++t) {
    if (t + 1 < NT) {
      issue_tile((t + 1) & 1, t + 1);
      // loads complete in order: <=5 outstanding => tile t's 5 loads landed
      asm volatile("s_wait_asynccnt 0x5" ::: "memory");
    } else {
      asm volatile("s_wait_asynccnt 0x0" ::: "memory");
    }
    const int buf = t & 1;
    float4 wv = stage[buf][0][tid];
    float4 cv = stage[buf][1][tid];
    float4 rv = stage[buf][2][tid];
    float4 mv = stage[buf][3][tid];
    float4 sv = stage[buf][4][tid];
    proc(wv.x, cv.x, rv.x, mv.x, sv.x);
    proc(wv.y, cv.y, rv.y, mv.y, sv.y);
    proc(wv.z, cv.z, rv.z, mv.z, sv.z);
    proc(wv.w, cv.w, rv.w, mv.w, sv.w);
  }

  // Block tree-reduce (plain sums + softmax-state merges)
  red[0][tid] = sum_w; red[1][tid] = ent; red[2][tid] = sc;
  red[3][tid] = tw;    red[4][tid] = nv;
  red[5][tid] = mt;    red[6][tid] = zt;  red[7][tid] = ct;
  red[8][tid] = ms;    red[9][tid] = zs;
  for (int off = 128; off > 0; off >>= 1) {
    __syncthreads();
    if (tid < off) {
      red[0][tid] += red[0][tid + off];
      red[1][tid] += red[1][tid + off];
      red[2][tid] += red[2][tid + off];
      red[3][tid] += red[3][tid + off];
      red[4][tid] += red[4][tid + off];
      // merge (mt, zt, ct)
      float m1 = red[5][tid],       z1 = red[6][tid],       c1 = red[7][tid];
      float m2 = red[5][tid + off], z2 = red[6][tid + off], c2 = red[7][tid + off];
      float mm = fmaxf(m1, m2);
      float a1 = (z1 > 0.f) ? __expf(m1 - mm) : 0.f;
      float a2 = (z2 > 0.f) ? __expf(m2 - mm) : 0.f;
      red[5][tid] = mm;
      red[6][tid] = z1 * a1 + z2 * a2;
      red[7][tid] = c1 * a1 + c2 * a2;
      // merge (ms, zs)
      m1 = red[8][tid];       z1 = red[9][tid];
      m2 = red[8][tid + off]; z2 = red[9][tid + off];
      mm = fmaxf(m1, m2);
      a1 = (z1 > 0.f) ? __expf(m1 - mm) : 0.f;
      a2 = (z2 > 0.f) ? __expf(m2 - mm) : 0.f;
      red[8][tid] = mm;
      red[9][tid] = z1 * a1 + z2 * a2;
    }
  }
  if (tid == 0) {
    float SW = red[0][0], ENT = red[1][0], SC = red[2][0], TW = red[3][0];
    float NV = red[4][0];
    float ZT = red[6][0], CT = red[7][0];
    float MS = red[8][0], ZS = red[9][0];
    ws[WS_INV + b] = 1.f / (SW + EPSF);
    ws[WS_ENT + b] = ENT;
    ws[WS_SC  + b] = SC;
    ws[WS_TW  + b] = TW;
    bool ok = (NV >= 2.f) && (ZT > 0.f) && (ZS > 0.f);
    // per_sample = max_s + log Z_s - E_target[s]   (sum of targets == 1)
    float per = ok ? (MS + __logf(ZS) - CT / ZT) / NV : 0.f;
    ws[WS_RK + b] = per;
    ws[WS_OK + b] = ok ? 1.f : 0.f;
  }
}

// ---------------------------------------------------------------------------
// Kernel 2a: deterministic column partials: partial[chunk][n] over 128 rows
// ---------------------------------------------------------------------------
__global__ __launch_bounds__(256) void col_partial_kernel(
    const float* __restrict__ weights, const float* __restrict__ masks,
    const float* __restrict__ inv, float* __restrict__ partial)
{
  const int n     = blockIdx.x * 256 + threadIdx.x;
  const int chunk = blockIdx.y;
  const int b0    = chunk * (BB / 32);
  float acc = 0.f;
#pragma unroll 4
  for (int r = 0; r < BB / 32; ++r) {
    const int b = b0 + r;
    const size_t idx = (size_t)b * NN + n;
    acc += weights[idx] * masks[idx] * inv[b];
  }
  partial[(size_t)chunk * NN + n] = acc;
}

// ---------------------------------------------------------------------------
// Kernel 2b: fold the 32 chunk-partials per column with WMMA:
//   colsum(1xN) = ones(1x32) x partial(32xN)
// Per wave: D(16x16) = A(16x4, ones) x B(4x16 tile of partial) + C,
// chained over 8 WMMAs (k-chunks 0..31). Every D row == column sums.
// B f32 4x16 lane layout mirrors the documented 16x4 f32 A layout:
//   VGPR0: lanes 0-15 = K0 row, lanes 16-31 = K2 row
//   VGPR1: lanes 0-15 = K1 row, lanes 16-31 = K3 row
// ---------------------------------------------------------------------------
__global__ __launch_bounds__(256) void col_sum_wmma_kernel(
    const float* __restrict__ partial, float* __restrict__ colsum)
{
  const int lane = threadIdx.x & 31;
  const int wave = threadIdx.x >> 5;
  const int n0   = blockIdx.x * 128 + wave * 16;   // 16 columns per wave
  const int half = lane >> 4;
  const int nIdx = lane & 15;

  v2f a;
  a.x = 1.f;
  a.y = 1.f;                       // A = ones(16x4)
  v8f c = {};                      // C/D accumulator
#pragma unroll
  for (int g = 0; g < 8; ++g) {
    const int k0 = g * 4 + 2 * half;
    v2f bm;
    bm.x = partial[(size_t)(k0    ) * NN + n0 + nIdx];
    bm.y = partial[(size_t)(k0 + 1) * NN + n0 + nIdx];
    c = __builtin_amdgcn_wmma_f32_16x16x4_f32(false, a, false, bm,
                                              (short)0, c, false, false);
  }
  if (lane < 16) colsum[n0 + nIdx] = c[0];
}

// ---------------------------------------------------------------------------
// Kernel 3: finalize everything into one scalar
// ---------------------------------------------------------------------------
__device__ __forceinline__ float block_reduce(float v, float* sbuf, int tid) {
  __syncthreads();
  sbuf[tid] = v;
  for (int off = 128; off > 0; off >>= 1) {
    __syncthreads();
    if (tid < off) sbuf[tid] += sbuf[tid + off];
  }
  __syncthreads();
  return sbuf[0];
}

__global__ __launch_bounds__(256) void final_kernel(
    const float* __restrict__ pr, const float* __restrict__ ws,
    const float* __restrict__ colsum, float* __restrict__ out)
{
  __shared__ float sbuf[256];
  const int tid = threadIdx.x;

  float aPr = 0.f, aPr2 = 0.f, aEnt = 0.f, aMse = 0.f, aRank = 0.f, aOk = 0.f;
  for (int b = tid; b < BB; b += 256) {
    float p = pr[b];
    aPr  += p;
    aPr2 += p * p;
    aEnt += ws[WS_ENT + b];
    float pc  = ws[WS_SC + b] / (ws[WS_TW + b] + EPSF);
    float tgt = 1.f / (1.f + __expf(-50.f * p));
    float d   = pc - tgt;
    aMse  += d * d;
    aRank += ws[WS_RK + b];
    aOk   += ws[WS_OK + b];
  }
  float aS = 0.f;
  constexpr float invB = 1.f / (float)BB;
  for (int n = tid; n < NN; n += 256) aS += colsum[n] * invB;

  float SPr  = block_reduce(aPr,  sbuf, tid);
  float SPr2 = block_reduce(aPr2, sbuf, tid);
  float SEnt = block_reduce(aEnt, sbuf, tid);
  float SMse = block_reduce(aMse, sbuf, tid);
  float SRk  = block_reduce(aRank, sbuf, tid);
  float SOk  = block_reduce(aOk, sbuf, tid);
  float S    = block_reduce(aS,  sbuf, tid);   // broadcast to all threads

  // entropy of normalized avg_sel
  float aB = 0.f;
  float invS = 1.f / (S + EPSF);
  for (int n = tid; n < NN; n += 256) {
    float p = colsum[n] * invB * invS;
    aB += p * __logf(p + EPSF);
  }
  float SB = block_reduce(aB, sbuf, tid);

  if (tid == 0) {
    float mean = SPr * invB;
    float var  = (SPr2 - (float)BB * mean * mean) / (float)(BB - 1);
    float sd   = sqrtf(fmaxf(var, 0.f));
    float sharpe = fminf(fmaxf(-mean / (sd + 0.01f), -10.f), 10.f);
    float conc = -SEnt * invB;                       // concentration loss
    float confL = SMse * invB;                       // confidence loss
    float aux = (SOk > 0.f) ? (SRk / SOk) : 0.f;     // ranking loss
    out[0] = 1.0f * (-mean)    // RETURN_W
           + 0.1f  * sharpe    // SHARPE_W
           + 0.01f * conc      // CONC_W
           + 0.1f  * confL     // CONF_W
           + 0.01f * SB        // BENT_W * (-batch_entropy)
           + 0.1f  * aux;      // AUX_W
  }
}

// ---------------------------------------------------------------------------
extern "C" void kernel_launch(void* const* d_in, const int* in_sizes, int n_in,
                              void* d_out, int out_size, void* d_ws, size_t ws_size,
                              hipStream_t stream) {
  (void)in_sizes; (void)n_in; (void)out_size; (void)ws_size;
  const float* pr = (const float*)d_in[0];
  const float* w  = (const float*)d_in[1];
  const float* c  = (const float*)d_in[2];
  const float* r  = (const float*)d_in[3];
  const float* m  = (const float*)d_in[4];
  const float* s  = (const float*)d_in[5];
  float* ws  = (float*)d_ws;
  float* out = (float*)d_out;

  row_kernel<<<BB, 256, 0, stream>>>(w, c, r, m, s, ws);
  col_partial_kernel<<<dim3(NN / 256, 32), 256, 0, stream>>>(w, m, ws + WS_INV, ws + WS_PART);
  col_sum_wmma_kernel<<<NN / 128, 256, 0, stream>>>(ws + WS_PART, ws + WS_COL);
  final_kernel<<<1, 256, 0, stream>>>(pr, ws, ws + WS_COL, out);
}